// TopKRouter_49572512530496
// MI455X (gfx1250) — compile-verified
//
#include <hip/hip_runtime.h>
#include <hip/hip_bf16.h>

// ---------------- problem constants (from reference) ----------------
#define NTOK   16384          // B*S = 4*4096 tokens
#define DDIM   4096           // hidden dim
#define NEXP   64             // experts
#define KTOP   8              // top-k
#define MOM    0.95f
#define BIASLR 0.01f

// ---------------- tiling ----------------
#define BM     32             // tokens per block
#define BK     64             // K chunk staged in LDS
#define LDX    (BK + 4)       // padded LDS row stride (floats): 68 -> conflict-free fragment reads
#define NBLK   (NTOK / BM)    // 512 blocks
#define NCH    (DDIM / BK)    // 64 K-chunks

// output layout (floats, concatenated in reference return order)
#define OFF_PROB  0
#define OFF_IDX   (NTOK * KTOP)            // 131072
#define OFF_Z     (2 * NTOK * KTOP)        // 262144
#define OFF_EMA   (OFF_Z + 1)              // 262145
#define OFF_BIAS  (OFF_EMA + NEXP)         // 262209

typedef float v2f  __attribute__((ext_vector_type(2)));
typedef float v8f  __attribute__((ext_vector_type(8)));

// ---- gfx1250 async global->LDS copy (bypasses VGPRs, tracked by ASYNCcnt) ----
__device__ __forceinline__ unsigned lds_off(const void* p) {
    // generic -> LDS(AS3) -> 32-bit LDS byte offset
    return (unsigned)(unsigned long long)(__attribute__((address_space(3))) const void*)p;
}

__device__ __forceinline__ void async_copy_b128(const void* gptr, void* lptr) {
    unsigned l = lds_off(lptr);
    unsigned long long g = (unsigned long long)gptr;
    asm volatile("global_load_async_to_lds_b128 %0, %1, off"
                 :: "v"(l), "v"(g) : "memory");
}

#define ASYNC_WAIT(n) asm volatile("s_wait_asynccnt " #n ::: "memory")

// -------------------------------------------------------------------
__global__ void router_init(int* __restrict__ g_hist) {
    if (threadIdx.x < NEXP) g_hist[threadIdx.x] = 0;
}

// -------------------------------------------------------------------
__global__ __launch_bounds__(256) void router_main(
    const float* __restrict__ x,        // [NTOK, DDIM]
    const float* __restrict__ Wg,       // [NEXP, DDIM]
    const float* __restrict__ bias,     // [NEXP]
    float* __restrict__ out_prob,       // [NTOK, KTOP]
    float* __restrict__ out_idx,        // [NTOK, KTOP] (indices stored as float)
    int*   __restrict__ g_hist,         // [NEXP]
    float* __restrict__ g_zpart)        // [NBLK]
{
    __shared__ __align__(16) float xs [2][BM][LDX];    // double-buffered x tile
    __shared__ __align__(16) float wsh[2][NEXP][LDX];  // double-buffered W tile
    __shared__ float lg[BM][NEXP + 1];                 // logits tile
    __shared__ float zarr[BM];
    __shared__ int   hist[NEXP];

    const int tid  = threadIdx.x;
    const int lane = tid & 31;
    const int wave = tid >> 5;
    const int wm   = wave >> 2;        // 0..1 : token sub-tile (16 rows)
    const int wn   = wave & 3;         // 0..3 : expert sub-tile (16 cols)
    const int half = lane >> 4;        // lane half selects K group per ISA layout
    const int l16  = lane & 15;
    const int t0   = blockIdx.x * BM;

    if (tid < NEXP) hist[tid] = 0;

    // per-thread staging coordinates: 16B granules, 16 granules per 64-float row
    const int srow = tid >> 4;          // 0..15
    const int scol = (tid & 15) << 2;   // 0,4,...,60

    // issue one chunk's async copies: 2 (x) + 4 (W) = 6 per thread
    auto stage = [&](int k0, int b) {
        #pragma unroll
        for (int r = 0; r < 2; ++r) {   // x: 32 rows x 64 floats
            const int row = srow + r * 16;
            async_copy_b128(x + (size_t)(t0 + row) * DDIM + k0 + scol,
                            &xs[b][row][scol]);
        }
        #pragma unroll
        for (int r = 0; r < 4; ++r) {   // W: 64 rows x 64 floats (L2-resident)
            const int row = srow + r * 16;
            async_copy_b128(Wg + (size_t)row * DDIM + k0 + scol,
                            &wsh[b][row][scol]);
        }
    };

    v8f acc = {};   // 16x16 f32 accumulator (8 VGPRs/lane, wave32)

    stage(0, 0);    // prologue: chunk 0 in flight

    for (int c = 0; c < NCH; ++c) {
        if (c + 1 < NCH) {
            stage((c + 1) * BK, (c + 1) & 1);   // keep HBM streaming
            ASYNC_WAIT(6);                      // oldest 6 (chunk c) complete
        } else {
            ASYNC_WAIT(0);
        }
        __syncthreads();

        const float* xrow = &xs [c & 1][wm * 16 + l16][0];   // A: M = l16
        const float* wrow = &wsh[c & 1][wn * 16 + l16][0];   // B: N = l16

        // f32 WMMA: A 16x4 = {K=2h, K=2h+1} per lane; B 4x16 identical float2 pattern.
        #pragma unroll
        for (int kk = 0; kk < BK; kk += 4) {
            v2f a = *(const v2f*)(xrow + kk + 2 * half);
            v2f b = *(const v2f*)(wrow + kk + 2 * half);
            acc = __builtin_amdgcn_wmma_f32_16x16x4_f32(
                false, a, false, b, (short)0, acc, false, false);
        }
        __syncthreads();    // protect buffer (c&1) before it is restaged
    }

    // ---- epilogue: +bias, spill logits tile to LDS (C layout: VGPR i -> M = i + 8*half) ----
    const float bn = bias[wn * 16 + l16];
    #pragma unroll
    for (int i = 0; i < 8; ++i)
        lg[wm * 16 + half * 8 + i][wn * 16 + l16] = acc[i] + bn;
    __syncthreads();

    // ---- per-token reduction: wave 0, one lane per token ----
    if (tid < BM) {
        const float* row = &lg[tid][0];
        float mx = row[0];
        #pragma unroll 8
        for (int e = 1; e < NEXP; ++e) mx = fmaxf(mx, row[e]);
        float s = 0.f;
        #pragma unroll 8
        for (int e = 0; e < NEXP; ++e) s += expf(row[e] - mx);
        const float lse = mx + logf(s);
        zarr[tid] = lse * lse;

        // top-8, tie-break = lowest index (matches lax.top_k)
        unsigned long long taken = 0ull;
        float vals[KTOP]; int idxs[KTOP];
        for (int j = 0; j < KTOP; ++j) {
            float bv = -3.4e38f; int bi = 0;
            for (int e = 0; e < NEXP; ++e) {
                if (!((taken >> e) & 1ull)) {
                    const float v = row[e];
                    if (v > bv) { bv = v; bi = e; }
                }
            }
            taken |= 1ull << bi;
            vals[j] = bv; idxs[j] = bi;
        }
        // softmax over top-8 (vals[0] is the max)
        float pr[KTOP], ss = 0.f;
        #pragma unroll
        for (int j = 0; j < KTOP; ++j) { pr[j] = expf(vals[j] - vals[0]); ss += pr[j]; }
        const float inv = 1.f / ss;
        const size_t base = (size_t)(t0 + tid) * KTOP;
        #pragma unroll
        for (int j = 0; j < KTOP; ++j) {
            out_prob[base + j] = pr[j] * inv;
            out_idx [base + j] = (float)idxs[j];
            atomicAdd(&hist[idxs[j]], 1);
        }
    }
    __syncthreads();

    if (tid < NEXP) atomicAdd(&g_hist[tid], hist[tid]);   // integer: deterministic
    if (tid == 0) {
        float z = 0.f;
        for (int i = 0; i < BM; ++i) z += zarr[i];
        g_zpart[blockIdx.x] = z;                           // write, not atomic -> deterministic
    }
}

// -------------------------------------------------------------------
__global__ void router_final(const int* __restrict__ g_hist,
                             const float* __restrict__ g_zpart,
                             const float* __restrict__ ema,
                             const float* __restrict__ bias,
                             float* __restrict__ out)
{
    const int t = threadIdx.x;
    const float inv_ntok = 1.0f / (float)NTOK;
    if (t < NEXP) {
        const float load = (float)g_hist[t] * inv_ntok;
        const float nema = MOM * ema[t] + (1.0f - MOM) * load;
        out[OFF_EMA  + t] = nema;
        out[OFF_BIAS + t] = bias[t] + BIASLR * (1.0f / (float)NEXP - nema);
    }
    if (t == 0) {
        float z = 0.f;
        for (int i = 0; i < NBLK; ++i) z += g_zpart[i];    // serial: deterministic
        out[OFF_Z] = z * inv_ntok;
    }
}

// -------------------------------------------------------------------
extern "C" void kernel_launch(void* const* d_in, const int* in_sizes, int n_in,
                              void* d_out, int out_size, void* d_ws, size_t ws_size,
                              hipStream_t stream) {
    const float* x    = (const float*)d_in[0];
    const float* Wg   = (const float*)d_in[1];
    const float* bias = (const float*)d_in[2];
    const float* ema  = (const float*)d_in[3];
    float* out = (float*)d_out;

    int*   g_hist  = (int*)d_ws;                 // 64 ints
    float* g_zpart = (float*)d_ws + NEXP;        // 512 floats

    router_init<<<1, 64, 0, stream>>>(g_hist);
    router_main<<<NBLK, 256, 0, stream>>>(x, Wg, bias,
                                          out + OFF_PROB, out + OFF_IDX,
                                          g_hist, g_zpart);
    router_final<<<1, 64, 0, stream>>>(g_hist, g_zpart, ema, bias, out);
}